// SSIM_13443247637111
// MI455X (gfx1250) — compile-verified
//
#include <hip/hip_runtime.h>
#include <math.h>

typedef __attribute__((ext_vector_type(2))) float v2f;
typedef __attribute__((ext_vector_type(8))) float v8f;

#define IMG   512
#define TILE  64
#define PAD   5
#define HALO  74            // TILE + 2*PAD
#define SIN   84            // LDS stride for input tiles (16B-aligned rows, 84%64=20)
#define COLOFF 3            // halo col c lives at LDS col c+3 (global 16B alignment)
#define NSEG  20            // 20 x b128 = 80 floats copied per row (cols 0..79)
#define HROWS 80            // horizontal-result rows (5 blocks of 16)
#define HSTR  66            // stride for H buffers (8*66 % 64 != 0 -> no bank clash)
#define HBUF  (HROWS*HSTR)

// D = A(16x4) * B(4x16) + C, full f32 (V_WMMA_F32_16X16X4_F32)
__device__ __forceinline__ v8f wmma4(v2f a, v2f b, v8f c) {
  return __builtin_amdgcn_wmma_f32_16x16x4_f32(false, a, false, b, (short)0, c,
                                               false, false);
}

// CDNA5 async global->LDS copy, 16 bytes per lane (ASYNCcnt-tracked)
__device__ __forceinline__ void async_copy_b128(unsigned lds_off,
                                                const float* gaddr) {
  asm volatile("global_load_async_to_lds_b128 %0, %1, off"
               :: "v"(lds_off), "v"(gaddr) : "memory");
}
__device__ __forceinline__ void async_wait0() {
  asm volatile("s_wait_asynccnt 0x0" ::: "memory");
}

// Fused masked-SSIM tile kernel. One 64x64 output tile per (b,c) per block.
__global__ __launch_bounds__(256) void ssim_tile_kernel(
    const float* __restrict__ img1, const float* __restrict__ img2,
    const float* __restrict__ match, float* __restrict__ partials) {
  extern __shared__ float smem[];
  float* s_in1 = smem;                       // 74*84 = 6216 floats
  float* s_in2 = smem + HALO * SIN;          // 6216 floats
  float* s_h   = smem + 2 * HALO * SIN;      // 5 * 80*66 = 26400 floats
  float* s_m   = s_h + HBUF;                 // match tile aliases h[1..2]

  const int tid   = threadIdx.x;
  const int lane  = tid & 31;
  // wave-uniform -> force into SGPR so subtile loops are scalar (EXEC stays all-1s
  // for WMMA instead of compiler-generated EXEC-masked vector loops)
  const int wave  = __builtin_amdgcn_readfirstlane(tid >> 5);
  const int i15   = lane & 15;
  const int khalf = (lane >> 4) << 1;        // 0 (lanes 0-15) or 2 (lanes 16-31)

  const int bz  = blockIdx.z;
  const int b   = bz / 3;
  const int ch  = bz % 3;
  const int tr0 = blockIdx.y * TILE;
  const int tc0 = blockIdx.x * TILE;

  // ---- per-thread 1D Gaussian (sigma = 1.5, 11 taps, normalized) ----
  float g[11];
  {
    float s = 0.f;
    for (int j = 0; j < 11; ++j) {
      float d = (float)(j - PAD);
      g[j] = __expf(-d * d / (2.f * 1.5f * 1.5f));
      s += g[j];
    }
    for (int j = 0; j < 11; ++j) g[j] /= s;
  }
  const float wbox = 1.0f / 121.0f;

  // ---- banded-weight chunks (serve as B in horizontal, A in vertical) ----
  v2f Wg[7], Wb[7];
#pragma unroll
  for (int kk = 0; kk < 7; ++kk) {
    int k0 = 4 * kk + khalf;
    unsigned d0 = (unsigned)(k0 - i15);
    unsigned d1 = (unsigned)(k0 + 1 - i15);
    Wg[kk].x = (d0 <= 10u) ? g[d0] : 0.f;
    Wg[kk].y = (d1 <= 10u) ? g[d1] : 0.f;
    Wb[kk].x = (d0 <= 10u) ? wbox : 0.f;
    Wb[kk].y = (d1 <= 10u) ? wbox : 0.f;
  }

  // ---- load halo tiles ----
  {
    const float* p1 = img1 + (size_t)bz * IMG * IMG;
    const float* p2 = img2 + (size_t)bz * IMG * IMG;
    const float* pm = match + (size_t)b * IMG * IMG;
    // rows tr0-5..tr0+68 ; async path copies cols tc0-8..tc0+71 (16B aligned)
    const bool interior = (tr0 >= PAD) && (tr0 + 69 <= IMG) &&
                          (tc0 >= 8) && (tc0 + 72 <= IMG);
    if (interior) {
      const int gr0 = tr0 - PAD;
      const int gc0 = tc0 - 8;               // multiple of 4 elements
      unsigned l1 = (unsigned)(size_t)s_in1;
      unsigned l2 = (unsigned)(size_t)s_in2;
      unsigned lm = (unsigned)(size_t)s_m;
      for (int idx = tid; idx < HALO * NSEG; idx += 256) {
        int rr = idx / NSEG, seg = idx - rr * NSEG;
        size_t gofs = (size_t)(gr0 + rr) * IMG + gc0 + seg * 4;
        unsigned lofs = (unsigned)((rr * SIN + seg * 4) * 4);
        async_copy_b128(l1 + lofs, p1 + gofs);
        async_copy_b128(l2 + lofs, p2 + gofs);
        async_copy_b128(lm + lofs, pm + gofs);
      }
      async_wait0();
    } else {
      for (int idx = tid; idx < HALO * HALO; idx += 256) {
        int rr = idx / HALO, cc = idx - rr * HALO;
        int gr = tr0 - PAD + rr, gc = tc0 - PAD + cc;
        bool ok = (gr >= 0) & (gr < IMG) & (gc >= 0) & (gc < IMG);
        size_t gofs = (size_t)(ok ? gr : 0) * IMG + (ok ? gc : 0);
        int o = rr * SIN + cc + COLOFF;
        s_in1[o] = ok ? p1[gofs] : 0.f;
        s_in2[o] = ok ? p2[gofs] : 0.f;
        s_m[o]   = ok ? pm[gofs] : 0.f;
      }
    }
  }
  __syncthreads();

  // ================= mask (box filter of match) =================
  for (int st = wave; st < 20; st += 8) {
    int RB = st >> 2, CB = st & 3;
    int r = 16 * RB + i15; if (r > HALO - 1) r = HALO - 1;
    int n0 = 16 * CB;
    v8f hm = {};
#pragma unroll
    for (int kk = 0; kk < 7; ++kk) {
      int c0 = n0 + 4 * kk + khalf;
      int c1 = c0 + 1;
      if (c0 > HALO - 1) c0 = HALO - 1;
      if (c1 > HALO - 1) c1 = HALO - 1;
      v2f a = { s_m[r * SIN + c0 + COLOFF], s_m[r * SIN + c1 + COLOFF] };
      hm = wmma4(a, Wb[kk], hm);
    }
#pragma unroll
    for (int i = 0; i < 8; ++i) {
      int row = 16 * RB + i + ((lane >> 4) << 3);
      s_h[row * HSTR + n0 + i15] = hm[i];
    }
  }
  __syncthreads();

  // vertical box -> accM (this wave's two 16x16 output subtiles)
  v8f accM[2];
  for (int q = 0; q < 2; ++q) {
    int st = wave + 8 * q;
    int vb = st >> 2, cb = st & 3;
    int r0 = vb << 4;
    int c0 = (cb << 4) + i15;
    v8f am = {};
#pragma unroll
    for (int kk = 0; kk < 7; ++kk) {
      int k0 = r0 + 4 * kk + khalf;        // <= 75 < HROWS, contents finite
      v2f bm = { s_h[k0 * HSTR + c0], s_h[(k0 + 1) * HSTR + c0] };
      am = wmma4(Wb[kk], bm, am);
    }
    accM[q] = am;
  }
  __syncthreads();   // s_h (incl. match alias) about to be overwritten

  // ================= joint horizontal Gaussian: 5 fields =================
  for (int st = wave; st < 20; st += 8) {
    int RB = st >> 2, CB = st & 3;
    int r = 16 * RB + i15; if (r > HALO - 1) r = HALO - 1;
    int n0 = 16 * CB;
    v8f h0 = {}, h1 = {}, h2 = {}, h3 = {}, h4 = {};
#pragma unroll
    for (int kk = 0; kk < 7; ++kk) {
      int c0 = n0 + 4 * kk + khalf;
      int c1 = c0 + 1;
      if (c0 > HALO - 1) c0 = HALO - 1;
      if (c1 > HALO - 1) c1 = HALO - 1;
      float x0 = s_in1[r * SIN + c0 + COLOFF], x1 = s_in1[r * SIN + c1 + COLOFF];
      float y0 = s_in2[r * SIN + c0 + COLOFF], y1 = s_in2[r * SIN + c1 + COLOFF];
      v2f a1  = { x0, x1 };
      v2f a2  = { y0, y1 };
      v2f a11 = { x0 * x0, x1 * x1 };
      v2f a22 = { y0 * y0, y1 * y1 };
      v2f a12 = { x0 * y0, x1 * y1 };
      h0 = wmma4(a1,  Wg[kk], h0);
      h1 = wmma4(a2,  Wg[kk], h1);
      h2 = wmma4(a11, Wg[kk], h2);
      h3 = wmma4(a22, Wg[kk], h3);
      h4 = wmma4(a12, Wg[kk], h4);
    }
#pragma unroll
    for (int i = 0; i < 8; ++i) {
      int row = 16 * RB + i + ((lane >> 4) << 3);
      int o = row * HSTR + n0 + i15;
      s_h[0 * HBUF + o] = h0[i];
      s_h[1 * HBUF + o] = h1[i];
      s_h[2 * HBUF + o] = h2[i];
      s_h[3 * HBUF + o] = h3[i];
      s_h[4 * HBUF + o] = h4[i];
    }
  }
  __syncthreads();

  // ============ vertical Gaussian + SSIM map + masked partials ============
  float p1 = 0.f, p2 = 0.f;
  for (int q = 0; q < 2; ++q) {
    int st = wave + 8 * q;
    int vb = st >> 2, cb = st & 3;
    int r0 = vb << 4;
    int c0 = (cb << 4) + i15;
    v8f a0 = {}, a1 = {}, a2 = {}, a3 = {}, a4 = {};
#pragma unroll
    for (int kk = 0; kk < 7; ++kk) {
      int k0 = r0 + 4 * kk + khalf;
      int o0 = k0 * HSTR + c0, o1 = (k0 + 1) * HSTR + c0;
      v2f b0 = { s_h[0 * HBUF + o0], s_h[0 * HBUF + o1] };
      v2f b1 = { s_h[1 * HBUF + o0], s_h[1 * HBUF + o1] };
      v2f b2 = { s_h[2 * HBUF + o0], s_h[2 * HBUF + o1] };
      v2f b3 = { s_h[3 * HBUF + o0], s_h[3 * HBUF + o1] };
      v2f b4 = { s_h[4 * HBUF + o0], s_h[4 * HBUF + o1] };
      a0 = wmma4(Wg[kk], b0, a0);
      a1 = wmma4(Wg[kk], b1, a1);
      a2 = wmma4(Wg[kk], b2, a2);
      a3 = wmma4(Wg[kk], b3, a3);
      a4 = wmma4(Wg[kk], b4, a4);
    }
    v8f am = accM[q];
#pragma unroll
    for (int i = 0; i < 8; ++i) {
      float m    = am[i] + 1e-7f;
      float mask = ((m > 0.5f) ? 1.f : 0.f) + 1e-7f;
      float mu1 = a0[i], mu2 = a1[i];
      float mu1s = mu1 * mu1, mu2s = mu2 * mu2, mu12 = mu1 * mu2;
      float sg1 = a2[i] - mu1s, sg2 = a3[i] - mu2s, sg12 = a4[i] - mu12;
      float num = (2.f * mu12 + 1e-4f) * (2.f * sg12 + 9e-4f);
      float den = (mu1s + mu2s + 1e-4f) * (sg1 + sg2 + 9e-4f);
      p1 += (1.f - num / den) * mask;
      if (ch == 0) p2 += mask;   // count mask once per (b,h,w)
    }
  }

  // ---- deterministic block reduction (reuse s_in1) ----
  __syncthreads();
  float* red = s_in1;
  red[tid] = p1;
  red[256 + tid] = p2;
  __syncthreads();
  for (int o = 128; o > 0; o >>= 1) {
    if (tid < o) {
      red[tid] += red[tid + o];
      red[256 + tid] += red[256 + tid + o];
    }
    __syncthreads();
  }
  if (tid == 0) {
    int bid = (bz * gridDim.y + blockIdx.y) * gridDim.x + blockIdx.x;
    partials[2 * bid]     = red[0];
    partials[2 * bid + 1] = red[256];
  }
}

__global__ __launch_bounds__(256) void ssim_finalize(
    const float* __restrict__ partials, int nblocks, float* __restrict__ out) {
  __shared__ float r1[256], r2[256];
  int t = threadIdx.x;
  float s1 = 0.f, s2 = 0.f;
  for (int i = t; i < nblocks; i += 256) {   // fixed order -> deterministic
    s1 += partials[2 * i];
    s2 += partials[2 * i + 1];
  }
  r1[t] = s1;
  r2[t] = s2;
  __syncthreads();
  for (int o = 128; o > 0; o >>= 1) {
    if (t < o) { r1[t] += r1[t + o]; r2[t] += r2[t + o]; }
    __syncthreads();
  }
  if (t == 0) out[0] = r1[0] / r2[0] / 3.0f;
}

extern "C" void kernel_launch(void* const* d_in, const int* in_sizes, int n_in,
                              void* d_out, int out_size, void* d_ws,
                              size_t ws_size, hipStream_t stream) {
  (void)in_sizes; (void)n_in; (void)out_size; (void)ws_size;
  const float* img1  = (const float*)d_in[0];
  const float* img2  = (const float*)d_in[1];
  const float* match = (const float*)d_in[2];
  float* out   = (float*)d_out;
  float* parts = (float*)d_ws;   // 3072 * 2 floats = 24.6 KB

  dim3 grid(IMG / TILE, IMG / TILE, 16 * 3);   // 8 x 8 x 48
  dim3 block(256);
  size_t shmem = (size_t)(2 * HALO * SIN + 5 * HBUF) * sizeof(float); // 155328 B
  ssim_tile_kernel<<<grid, block, shmem, stream>>>(img1, img2, match, parts);
  ssim_finalize<<<1, 256, 0, stream>>>(parts, 8 * 8 * 48, out);
}